// SimpleLossCompute_67044439490661
// MI455X (gfx1250) — compile-verified
//
#include <hip/hip_runtime.h>
#include <stdint.h>

// Problem constants (from the reference module)
static constexpr int   kNClauses = 262144;
static constexpr float kP = 10.0f;   // smooth-max temperature
static constexpr float kA = 10.0f;   // push_to_side sharpness

// ---------------------------------------------------------------------------
// CDNA5 async Global -> LDS copy (ASYNCcnt-tracked), non-temporal:
// the adjacency stream is 256 MB read exactly once, so NT keeps the hot
// xv/numer/denom working set (~3 MB) resident in the 192 MB L2.
// global_load_async_to_lds_b128  vdst(lds_byte_addr), vaddr(32b offset), saddr(base)
// ---------------------------------------------------------------------------
__device__ __forceinline__ void async_ld16_nt(uint32_t lds_off, int voff, const void* sbase) {
    asm volatile("global_load_async_to_lds_b128 %0, %1, %2 th:TH_LOAD_NT"
                 :: "v"(lds_off), "v"(voff), "s"(sbase)
                 : "memory");
}

__device__ __forceinline__ uint32_t lds_offset_of(const void* p) {
    // Flat LDS addresses are {SHARED_BASE_hi32, lds_offset32}; truncation yields
    // the workgroup-relative LDS byte offset used by DS / async-LDS instructions.
    return (uint32_t)(uintptr_t)p;
}

// ---------------------------------------------------------------------------
// Phase 1: scatter  numer[c] += x*exp(P*x),  denom[c] += exp(P*x)
// Adjacency (256 MB of int64) is streamed via double-buffered async-to-LDS.
// Each wave owns a private LDS region: [2 buffers][4 streams][64 int64].
// streams: 0 = pos clause, 1 = pos var, 2 = neg clause, 3 = neg var
// One tile = 64 literals per polarity per wave (2 per lane, B128 per lane/stream).
// ---------------------------------------------------------------------------
__global__ void __launch_bounds__(256)
sat_scatter_kernel(const long long* __restrict__ adj_pos,
                   const long long* __restrict__ adj_neg,
                   const float*     __restrict__ xv,
                   float*           __restrict__ numer,
                   float*           __restrict__ denom,
                   int E, int ntiles)
{
    __shared__ long long sm[8][2][4][64];   // 32 KB per workgroup

    const int lane = threadIdx.x & 31;
    const int wave = threadIdx.x >> 5;
    const int gw   = blockIdx.x * 8 + wave;      // global wave id (uniform per wave)
    const int W    = gridDim.x * 8;              // total waves

    const char* pc = (const char*)adj_pos;           // pos clause ids
    const char* pv = (const char*)(adj_pos + E);     // pos var ids
    const char* nc = (const char*)adj_neg;           // neg clause ids
    const char* nv = (const char*)(adj_neg + E);     // neg var ids

    auto issue = [&](int t, int b) {
        // byte offset of this lane's 2 literals within each int64 stream
        const int voff = t * 512 + lane * 16;        // 64 lits * 8 B per tile
        async_ld16_nt(lds_offset_of(&sm[wave][b][0][lane * 2]), voff, pc);
        async_ld16_nt(lds_offset_of(&sm[wave][b][1][lane * 2]), voff, pv);
        async_ld16_nt(lds_offset_of(&sm[wave][b][2][lane * 2]), voff, nc);
        async_ld16_nt(lds_offset_of(&sm[wave][b][3][lane * 2]), voff, nv);
    };

    int buf = 0;
    if (gw < ntiles) issue(gw, buf);

    for (int t = gw; t < ntiles; t += W) {
        const int tn = t + W;
        if (tn < ntiles) {
            issue(tn, buf ^ 1);                       // prefetch next tile (4 ops)
            asm volatile("s_wait_asynccnt 4" ::: "memory");  // current tile landed
        } else {
            asm volatile("s_wait_asynccnt 0" ::: "memory");
        }

#pragma unroll
        for (int j = 0; j < 2; ++j) {
            const int idx = lane * 2 + j;
            // positive literal
            const int c0 = (int)sm[wave][buf][0][idx];
            const int v0 = (int)sm[wave][buf][1][idx];
            const float x0 = xv[v0];                  // L2-resident gather (1 MB)
            const float e0 = __expf(kP * x0);
            atomicAdd(&numer[c0], x0 * e0);           // global_atomic_add_f32, L2
            atomicAdd(&denom[c0], e0);
            // negated literal
            const int c1 = (int)sm[wave][buf][2][idx];
            const int v1 = (int)sm[wave][buf][3][idx];
            const float x1 = 1.0f - xv[v1];
            const float e1 = __expf(kP * x1);
            atomicAdd(&numer[c1], x1 * e1);
            atomicAdd(&denom[c1], e1);
        }
        buf ^= 1;
    }

    // Tail (E not a multiple of 64) — empty for E = 8388608, kept for safety.
    const int tail0 = ntiles * 64;
    for (int i = tail0 + blockIdx.x * blockDim.x + threadIdx.x; i < E;
         i += gridDim.x * blockDim.x) {
        const int c0 = (int)adj_pos[i];
        const int v0 = (int)adj_pos[E + i];
        const float x0 = xv[v0];
        const float e0 = __expf(kP * x0);
        atomicAdd(&numer[c0], x0 * e0);
        atomicAdd(&denom[c0], e0);
        const int c1 = (int)adj_neg[i];
        const int v1 = (int)adj_neg[E + i];
        const float x1 = 1.0f - xv[v1];
        const float e1 = __expf(kP * x1);
        atomicAdd(&numer[c1], x1 * e1);
        atomicAdd(&denom[c1], e1);
    }
}

// ---------------------------------------------------------------------------
// Phase 2: loss = sum over clauses of softplus(A*(0.5 - numer/denom))
//   -log(sigmoid(A*(m-0.5))) = log(1 + exp(A*(0.5-m)))
// wave32 shuffle reduction + one float atomic per wave.
// ---------------------------------------------------------------------------
__global__ void __launch_bounds__(256)
sat_reduce_kernel(const float* __restrict__ numer,
                  const float* __restrict__ denom,
                  float* __restrict__ out, int nclauses)
{
    float acc = 0.0f;
    for (int i = blockIdx.x * blockDim.x + threadIdx.x; i < nclauses;
         i += gridDim.x * blockDim.x) {
        const float m = numer[i] / denom[i];
        const float z = kA * (0.5f - m);             // |z| <= ~5 since m in (0,1)
        acc += log1pf(__expf(z));                    // stable softplus
    }
#pragma unroll
    for (int off = 16; off > 0; off >>= 1)
        acc += __shfl_down(acc, off, 32);
    if ((threadIdx.x & 31) == 0)
        atomicAdd(out, acc);
}

// ---------------------------------------------------------------------------
// Launch: zero workspace+output (capture-legal memsets), scatter, reduce.
// ---------------------------------------------------------------------------
extern "C" void kernel_launch(void* const* d_in, const int* in_sizes, int n_in,
                              void* d_out, int out_size, void* d_ws, size_t ws_size,
                              hipStream_t stream) {
    (void)n_in; (void)out_size; (void)ws_size;
    const float*     xv      = (const float*)d_in[0];
    const long long* adj_pos = (const long long*)d_in[1];
    const long long* adj_neg = (const long long*)d_in[2];

    const int E = in_sizes[1] / 2;   // literals per polarity (adj is 2 x E)

    float* numer = (float*)d_ws;
    float* denom = numer + kNClauses;

    hipMemsetAsync(d_ws, 0, 2 * (size_t)kNClauses * sizeof(float), stream);
    hipMemsetAsync(d_out, 0, sizeof(float), stream);

    const int ntiles = E / 64;
    sat_scatter_kernel<<<1024, 256, 0, stream>>>(adj_pos, adj_neg, xv,
                                                 numer, denom, E, ntiles);
    sat_reduce_kernel<<<256, 256, 0, stream>>>(numer, denom, (float*)d_out,
                                               kNClauses);
}